// FairnessLoss_56281251447380
// MI455X (gfx1250) — compile-verified
//
#include <hip/hip_runtime.h>
#include <stdint.h>

// Problem constants (match the reference).
#define T_DIM 1000
#define B_DIM 64
#define V_DIM 1024
#define LMAX  64
#define S_DIM (2 * LMAX + 1)   // 129 extended states
#define NEGV  (-1.0e30f)
#define LAMBDA_FAIR 0.5f

// ---- CDNA5 async global->LDS staging (guarded; safe fallback) -------------
#if defined(__has_builtin)
#  if __has_builtin(__builtin_amdgcn_global_load_async_to_lds_b128) && \
      __has_builtin(__builtin_amdgcn_s_wait_asynccnt)
#    define USE_ASYNC_LDS 1
#  endif
#endif

// Builtin signature (from hipcc diagnostic): arg0 is
// '__attribute__((__vector_size__(4*sizeof(int)))) int __device__ *'
// i.e. a GCC-vector int4 pointer in the global (AS1) address space; arg1 is
// the LDS (AS3) counterpart.
typedef int v4i __attribute__((vector_size(16)));
typedef __attribute__((address_space(1))) v4i GV4;
typedef __attribute__((address_space(3))) v4i LV4;

// Copy 16 bytes from global to LDS. Async path uses the CDNA5
// GLOBAL_LOAD_ASYNC_TO_LDS_B128 (ASYNCcnt-tracked); fallback is a plain
// b128 load + ds_store_b128.
__device__ __forceinline__ void stage16(const float* gsrc, float* ldst) {
#ifdef USE_ASYNC_LDS
  __builtin_amdgcn_global_load_async_to_lds_b128(
      (GV4*)(uintptr_t)gsrc,
      (LV4*)(uint32_t)(uintptr_t)ldst,   // low 32 bits = LDS byte offset
      /*offset=*/0, /*cpol=*/0);
#else
  *(float4*)ldst = *(const float4*)gsrc;
#endif
}

__device__ __forceinline__ void stage_wait() {
#ifdef USE_ASYNC_LDS
  __builtin_amdgcn_s_wait_asynccnt(0);  // s_wait_asynccnt 0
#endif
}

// log(exp(a)+exp(b)), safe for NEGV sentinels (no inf/nan generated).
__device__ __forceinline__ float lax(float a, float b) {
  float m = fmaxf(a, b);
  float d = fminf(a, b) - m;          // <= 0; NEGV-NEGV -> 0 (fine)
  return m + log1pf(__expf(d));
}

// One block per batch sample. 256 threads: threads [0,129) own alpha states,
// all 256 cooperatively stage the 4KB log_probs row (256 * 16B) each step.
__global__ __launch_bounds__(256, 1) void ctc_alpha_kernel(
    const float* __restrict__ lp,       // [T, B, V]
    const int* __restrict__ targets,    // [B*LMAX] flat
    const int* __restrict__ in_len,     // [B]
    const int* __restrict__ tg_len,     // [B]
    float* __restrict__ per_sample)     // [B] out
{
  __shared__ __align__(16) float row[2][V_DIM];     // double-buffered em rows
  __shared__ float abuf[2][S_DIM + 2];              // alpha, 2-pad at front

  const int b   = blockIdx.x;
  const int tid = threadIdx.x;
  const int il  = in_len[b];
  const int tl  = tg_len[b];

  // exclusive cumsum offset into flat targets (uniform scalar loop, B<=64)
  int offset = 0;
  for (int i = 0; i < b; ++i) offset += tg_len[i];

  const int  s      = tid;
  const bool active = (s < S_DIM);
  int  lab   = 0;      // ext[s]: blanks at even s, labels at odd s
  bool allow = false;  // allow_skip[s]
  if (active && (s & 1)) {
    const int j = (s - 1) >> 1;
    if (j < tl) {
      lab = targets[offset + j];                       // in [1,V)
      if (j >= 1) allow = (lab != targets[offset + j - 1]);
    }
  }

  if (tid < 2) {                  // NEG pads, written once for both buffers
    abuf[0][tid] = NEGV;
    abuf[1][tid] = NEGV;
  }

  // ---- stage row t=0 and initialize alpha ----
  stage16(lp + (size_t)b * V_DIM + tid * 4, &row[0][tid * 4]);
  stage_wait();
  __syncthreads();

  if (active) {
    float a = NEGV;
    if (s == 0)                  a = row[0][0];        // em[0, blank]
    else if (s == 1 && tl > 0)   a = row[0][lab];      // em[0, first label]
    abuf[0][2 + s] = a;
  }
  if (il > 1)
    stage16(lp + ((size_t)B_DIM + b) * V_DIM + tid * 4, &row[1][tid * 4]);
  stage_wait();
  __syncthreads();

  // ---- alpha recursion, one barrier per timestep (double-buffered) ----
  int pa = 0;
  for (int t = 1; t < il; ++t) {
    const int rb = t & 1;          // buffer holding row t
    const int qa = pa ^ 1;
    if (t + 1 < il)                // pipeline: stage row t+1 during compute
      stage16(lp + ((size_t)(t + 1) * B_DIM + b) * V_DIM + tid * 4,
              &row[rb ^ 1][tid * 4]);
    if (t + 2 < il)                // gfx1250 global_prefetch_b8, 2 rows ahead
      __builtin_prefetch(lp + ((size_t)(t + 2) * B_DIM + b) * V_DIM + tid * 4,
                         0, 1);
    if (active) {
      const float a0 = abuf[pa][2 + s];                  // alpha[s]
      const float a1 = abuf[pa][1 + s];                  // alpha[s-1]
      const float a2 = allow ? abuf[pa][s] : NEGV;       // alpha[s-2]
      const float em = row[rb][lab];
      abuf[qa][2 + s] = lax(lax(a0, a1), a2) + em;
    }
    stage_wait();
    __syncthreads();
    pa = qa;
  }

  // ---- end_ll at t = il-1 (reference freezes `final` there) ----
  if (tid == 0) {
    const int   last = 2 * tl;
    const float a1   = abuf[pa][2 + last];
    const float a2   = (tl > 0) ? abuf[pa][1 + last] : NEGV;
    float loss = (tl > 0) ? -lax(a1, a2) : 0.0f;
    if (!(loss < 1e29f)) loss = 0.0f;                  // zero_infinity
    per_sample[b] = loss;
  }
}

// Tiny reduction: base mean, 2 group means, unbiased variance, total.
__global__ void fairness_reduce_kernel(const float* __restrict__ per_sample,
                                       const int* __restrict__ group_ids,
                                       float* __restrict__ out) {
  if (threadIdx.x == 0 && blockIdx.x == 0) {
    float sum = 0.0f;
    float gsum[2] = {0.0f, 0.0f};
    float gcnt[2] = {0.0f, 0.0f};
    for (int i = 0; i < B_DIM; ++i) {
      const float v = per_sample[i];
      const int   g = group_ids[i] & 1;
      sum += v;
      gsum[g] += v;
      gcnt[g] += 1.0f;
    }
    const float base = sum / (float)B_DIM;
    const float m0 = gsum[0] / fmaxf(gcnt[0], 1.0f);
    const float m1 = gsum[1] / fmaxf(gcnt[1], 1.0f);
    const float mg = 0.5f * (m0 + m1);
    const float var = (m0 - mg) * (m0 - mg) + (m1 - mg) * (m1 - mg); // /(2-1)
    out[0] = base + LAMBDA_FAIR * var;  // total
    out[1] = base;                      // base_loss
    out[2] = var;                       // fairness_penalty
  }
}

extern "C" void kernel_launch(void* const* d_in, const int* in_sizes, int n_in,
                              void* d_out, int out_size, void* d_ws, size_t ws_size,
                              hipStream_t stream) {
  const float* lp      = (const float*)d_in[0];  // [T*B*V] f32
  const int*   targets = (const int*)d_in[1];    // [B*LMAX] i32
  const int*   in_len  = (const int*)d_in[2];    // [B] i32
  const int*   tg_len  = (const int*)d_in[3];    // [B] i32
  const int*   gid     = (const int*)d_in[4];    // [B] i32
  float* per_sample = (float*)d_ws;              // 64 floats of scratch
  float* out        = (float*)d_out;             // [total, base, penalty]

  ctc_alpha_kernel<<<B_DIM, 256, 0, stream>>>(lp, targets, in_len, tg_len,
                                              per_sample);
  fairness_reduce_kernel<<<1, 32, 0, stream>>>(per_sample, gid, out);
}